// CellList_38250978738343
// MI455X (gfx1250) — compile-verified
//
#include <hip/hip_runtime.h>
#include <hip/hip_bf16.h>
#include <math.h>

// ---------------------------------------------------------------------------
// Cell-list neighbor pairs on MI455X (gfx1250).
//   d2(i,j) = |ri|^2 + |qj|^2 - 2 ri.qj ,  qj = rj - s*diag
// One block = one 16-row i-tile. SoA coords + norms (128 KB) are staged into
// LDS once per block with GLOBAL_LOAD_ASYNC_TO_LDS_B128 (ASYNCcnt), then each
// wave sweeps 64 j-tiles x 14 shifts, one V_WMMA_F32_16X16X4_F32 per tile.
// The WMMA computes d2 directly:  D = (-2*ri) . qj + (ni + nq).
// ---------------------------------------------------------------------------

typedef __attribute__((ext_vector_type(2))) float v2f;
typedef __attribute__((ext_vector_type(8))) float v8f;

#define N_ATOMS   8192
#define TILE      16
#define NTILES    (N_ATOMS / TILE)     // 512
#define NSHIFT    14
#define WAVES_PB  8                    // 8 wave32s per block
#define THREADS   (32 * WAVES_PB)
#define CUTOFF2   (5.2f * 5.2f)
#define LDS_FLOATS (4 * N_ATOMS)       // sx | sy | sz | n2
#define LDS_BYTES  (LDS_FLOATS * 4)    // 128 KB

__constant__ int c_shift[NSHIFT][3] = {
    { 0,  0,  0},
    {-1,  0,  0}, {-1, -1,  0}, { 0, -1,  0}, { 1, -1,  0},
    {-1,  1, -1}, { 0,  1, -1}, { 1,  1, -1}, {-1,  0, -1},
    { 0,  0, -1}, { 1,  0, -1}, {-1, -1, -1}, { 0, -1, -1}, { 1, -1, -1}};

// ---------------------------------------------------------------------------
// Kernel 1: wrap coordinates into the central cell, store SoA + squared norms.
// ---------------------------------------------------------------------------
__global__ void cell_prep_kernel(const float* __restrict__ coords,
                                 const float* __restrict__ cell,
                                 float* __restrict__ ws) {
    int i = blockIdx.x * blockDim.x + threadIdx.x;
    if (i >= N_ATOMS) return;
    float bx = cell[0], by = cell[4], bz = cell[8];
    float x = coords[3 * i + 0];
    float y = coords[3 * i + 1];
    float z = coords[3 * i + 2];
    x -= floorf(x / bx) * bx;
    y -= floorf(y / by) * by;
    z -= floorf(z / bz) * bz;
    ws[i]               = x;
    ws[N_ATOMS + i]     = y;
    ws[2 * N_ATOMS + i] = z;
    ws[3 * N_ATOMS + i] = x * x + y * y + z * z;
}

// ---------------------------------------------------------------------------
// Kernel 2: grid = (NTILES), block = 256. LDS-resident O(N^2) tile sweep.
// ---------------------------------------------------------------------------
__global__ void __launch_bounds__(THREADS)
cell_pairs_kernel(const float* __restrict__ ws,     // sx|sy|sz|n2, 4*N floats
                  const float* __restrict__ cell,
                  float* __restrict__ out, int P,
                  unsigned int* __restrict__ counter) {
    extern __shared__ float smem[];                 // LDS_FLOATS floats

    const int tid  = threadIdx.x;
    const int lane = tid & 31;
    const int half = lane >> 4;      // 0: K=0,1   1: K=2,3 (ISA 16x4 f32 layout)
    const int lo   = lane & 15;
    const int w    = tid >> 5;       // wave id in block

    // ---- Stage the whole SoA working set into LDS (async DMA path). -------
    // 32768 floats = 8192 x b128 chunks; 256 threads -> 32 chunks each.
    {
        const unsigned lds0 = (unsigned)(uintptr_t)smem;   // low 32 bits = LDS offset
        for (int c = tid; c < LDS_FLOATS / 4; c += THREADS) {
            unsigned laddr = lds0 + (unsigned)c * 16u;
            unsigned long long ga = (unsigned long long)(uintptr_t)(ws + 4 * c);
            asm volatile("global_load_async_to_lds_b128 %0, %1, off"
                         :: "v"(laddr), "v"(ga) : "memory");
        }
        asm volatile("s_wait_asynccnt 0x0" ::: "memory");
    }
    __syncthreads();

    const int it = blockIdx.x;
    const int ib = it * TILE;

    // ---- Per-wave invariants: A operand (scaled by -2) and row norms. -----
    const int im = ib + lo;
    const float aix = smem[im];
    const float aiy = smem[N_ATOMS + im];
    const float aiz = smem[2 * N_ATOMS + im];
    v2f a;
    a.x = -2.0f * ((half == 0) ? aix : aiz);
    a.y = (half == 0) ? (-2.0f * aiy) : 0.0f;

    v8f niv;
#pragma unroll
    for (int v = 0; v < 8; ++v)
        niv[v] = smem[3 * N_ATOMS + ib + v + 8 * half];   // |ri|^2, rows v+8*half

    // Shift vectors in length units.
    const float d0 = cell[0], d1 = cell[4], d2c = cell[8];
    float shx[NSHIFT], shy[NSHIFT], shz[NSHIFT];
#pragma unroll
    for (int k = 0; k < NSHIFT; ++k) {
        shx[k] = (float)c_shift[k][0] * d0;
        shy[k] = (float)c_shift[k][1] * d1;
        shz[k] = (float)c_shift[k][2] * d2c;
    }

    // ---- Sweep j tiles (wave-strided) x 14 shifts. -------------------------
    for (int jt = w; jt < NTILES; jt += WAVES_PB) {
        const int jb = jt * TILE;
        const int jn = jb + lo;
        const float cjx = smem[jn];
        const float cjy = smem[N_ATOMS + jn];
        const float cjz = smem[2 * N_ATOMS + jn];
        const int j = jn;

        for (int k = 0; k < NSHIFT; ++k) {
            if (k == 0 && it > jt) continue;   // wave-uniform triangle skip

            const float qx = cjx - shx[k];
            const float qy = cjy - shy[k];
            const float qz = cjz - shz[k];
            v2f b;
            b.x = (half == 0) ? qx : qz;
            b.y = (half == 0) ? qy : 0.0f;
            const float nq = qx * qx + qy * qy + qz * qz;

            v8f cin;
#pragma unroll
            for (int v = 0; v < 8; ++v) cin[v] = niv[v] + nq;

            // D = (-2*ri) . qj + (|ri|^2 + |qj|^2) = d2, one 16x16 tile.
            v8f d2v = __builtin_amdgcn_wmma_f32_16x16x4_f32(
                /*neg_a=*/false, a, /*neg_b=*/false, b,
                /*c_mod=*/(short)0, cin, /*reuse_a=*/false, /*reuse_b=*/false);

            const float shf0 = (float)c_shift[k][0];
            const float shf1 = (float)c_shift[k][1];
            const float shf2 = (float)c_shift[k][2];

#pragma unroll
            for (int v = 0; v < 8; ++v) {
                const int i = ib + v + 8 * half;
                bool pass = d2v[v] <= CUTOFF2;
                if (k == 0) pass = pass && (i < j);
                if (pass) {
                    const unsigned int pos = atomicAdd(counter, 1u);
                    if (pos < (unsigned int)P) {
                        out[pos]     = (float)i;          // atom_pairs row 0
                        out[P + pos] = (float)j;          // atom_pairs row 1
                        float* sh = out + 2 * P + 3 * (int)pos;
                        sh[0] = shf0; sh[1] = shf1; sh[2] = shf2;
                    }
                }
            }
        }
    }
}

// ---------------------------------------------------------------------------
// Host launch.  d_in order: species(int64), coordinates(f32), cell(f32), pbc.
// d_ws layout: sx | sy | sz | n2  (N floats each) | counter (u32).
// ---------------------------------------------------------------------------
extern "C" void kernel_launch(void* const* d_in, const int* in_sizes, int n_in,
                              void* d_out, int out_size, void* d_ws, size_t ws_size,
                              hipStream_t stream) {
    const float* coords = (const float*)d_in[1];
    const float* cell   = (const float*)d_in[2];

    float* ws = (float*)d_ws;
    unsigned int* counter = (unsigned int*)(ws + 4 * N_ATOMS);

    hipMemsetAsync(counter, 0, sizeof(unsigned int), stream);

    cell_prep_kernel<<<(N_ATOMS + 255) / 256, 256, 0, stream>>>(coords, cell, ws);

    const int P = out_size / 5;   // out = [i row | j row | 3-wide shifts]
    cell_pairs_kernel<<<dim3(NTILES), dim3(THREADS), LDS_BYTES, stream>>>(
        ws, cell, (float*)d_out, P, counter);
}